// GraphLayer_52802327937707
// MI455X (gfx1250) — compile-verified
//
#include <hip/hip_runtime.h>

typedef float v2f __attribute__((ext_vector_type(2)));
typedef float v8f __attribute__((ext_vector_type(8)));

#define D_DIM 64

// ---------------------------------------------------------------------------
// Degree kernels: deg starts at 1.0 (self loop), +1 per incoming edge, then
// dinv = rsqrt(deg) in place.
// ---------------------------------------------------------------------------
__global__ void k_init_deg(float* __restrict__ deg, int n) {
    int i = blockIdx.x * blockDim.x + threadIdx.x;
    if (i < n) deg[i] = 1.0f;
}

__global__ void k_deg(const int* __restrict__ ei, float* __restrict__ deg, int e) {
    int t = blockIdx.x * blockDim.x + threadIdx.x;
    if (t < e) atomicAdd(&deg[ei[e + t]], 1.0f);   // dst row of edge_index[2,E]
}

__global__ void k_dinv(float* __restrict__ deg, int n) {
    int i = blockIdx.x * blockDim.x + threadIdx.x;
    if (i < n) deg[i] = rsqrtf(deg[i]);
}

// ---------------------------------------------------------------------------
// h = x @ W^T via V_WMMA_F32_16X16X4_F32 (full fp32 precision).
// Each wave: one 16-row strip, all 64 output cols (4 N-tiles of 16).
// A 16x4 layout: lanes 0-15 hold K={k,k+1}, lanes 16-31 hold K={k+2,k+3}.
// B 4x16 layout mirrors C/D striping: VGPR v holds K = v + 2*(lane>=16).
// Epilogue: h -> ws, and seed out = h*dinv(row)^2 + bias (self-loop + b).
// ---------------------------------------------------------------------------
__global__ void __launch_bounds__(256) k_gemm(const float* __restrict__ x,
                                              const float* __restrict__ W,
                                              const float* __restrict__ bias,
                                              const float* __restrict__ dinv,
                                              float* __restrict__ h,
                                              float* __restrict__ out,
                                              int n) {
    __shared__ float Wlds[D_DIM * D_DIM];
    // cooperative 16KB copy of W (row-major [out=64][in=64])
    for (int i = threadIdx.x; i < D_DIM * D_DIM / 4; i += 256)
        ((float4*)Wlds)[i] = ((const float4*)W)[i];
    __syncthreads();

    const int wave  = threadIdx.x >> 5;
    const int lane  = threadIdx.x & 31;
    const int strip = blockIdx.x * 8 + wave;
    const int rbase = strip * 16;
    if (rbase >= n) return;

    const int hi = lane >> 4;     // 0: K lo-pair, 1: K hi-pair
    const int lo = lane & 15;     // M (for A) / N (for B)

    const float* xrow = x + (size_t)(rbase + lo) * D_DIM;

    v8f acc[4];
    acc[0] = (v8f){}; acc[1] = (v8f){}; acc[2] = (v8f){}; acc[3] = (v8f){};

    for (int k = 0; k < D_DIM; k += 4) {
        const int kk = k + 2 * hi;
        const v2f a = *(const v2f*)(xrow + kk);            // {x[row,kk], x[row,kk+1]}
#pragma unroll
        for (int t = 0; t < 4; ++t) {
            // B[kl][n] = W^T tile = W[n][kl]; contiguous pair in W's row
            const v2f bf = *(const v2f*)(&Wlds[(t * 16 + lo) * D_DIM + kk]);
            acc[t] = __builtin_amdgcn_wmma_f32_16x16x4_f32(
                false, a, false, bf, (short)0, acc[t], false, false);
        }
    }

    // C/D layout: VGPR r -> row = rbase + r + 8*hi, col = t*16 + lo
#pragma unroll
    for (int r = 0; r < 8; ++r) {
        const int row = rbase + r + 8 * hi;
        const float di = dinv[row];
        const float slnorm = di * di;            // self-loop norm dinv^2
        const size_t base = (size_t)row * D_DIM;
#pragma unroll
        for (int t = 0; t < 4; ++t) {
            const int col = t * 16 + lo;
            const float v = acc[t][r];
            h[base + col]   = v;
            out[base + col] = v * slnorm + bias[col];
        }
    }
}

// ---------------------------------------------------------------------------
// Edge scatter: 16 threads per edge, float4 gather of h[src], 4 f32 atomics
// into out[dst]. norm = dinv[src]*dinv[dst].
// ---------------------------------------------------------------------------
__global__ void k_edge(const int* __restrict__ ei,
                       const float* __restrict__ h,
                       const float* __restrict__ dinv,
                       float* __restrict__ out, int e) {
    int t = blockIdx.x * blockDim.x + threadIdx.x;
    int idx = t >> 4;
    if (idx >= e) return;
    int part = (t & 15) * 4;
    int s = ei[idx];
    int d = ei[e + idx];
    float norm = dinv[s] * dinv[d];
    float4 hv = *(const float4*)(h + (size_t)s * D_DIM + part);
    float* dst = out + (size_t)d * D_DIM + part;
    atomicAdd(dst + 0, hv.x * norm);
    atomicAdd(dst + 1, hv.y * norm);
    atomicAdd(dst + 2, hv.z * norm);
    atomicAdd(dst + 3, hv.w * norm);
}

// ---------------------------------------------------------------------------
// out = relu(out + x), vectorized float4.
// ---------------------------------------------------------------------------
__global__ void k_final(const float* __restrict__ x, float* __restrict__ out, int n4) {
    int i = blockIdx.x * blockDim.x + threadIdx.x;
    if (i >= n4) return;
    float4 a = ((const float4*)out)[i];
    float4 b = ((const float4*)x)[i];
    a.x = fmaxf(a.x + b.x, 0.0f);
    a.y = fmaxf(a.y + b.y, 0.0f);
    a.z = fmaxf(a.z + b.z, 0.0f);
    a.w = fmaxf(a.w + b.w, 0.0f);
    ((float4*)out)[i] = a;
}

extern "C" void kernel_launch(void* const* d_in, const int* in_sizes, int n_in,
                              void* d_out, int out_size, void* d_ws, size_t ws_size,
                              hipStream_t stream) {
    const float* x    = (const float*)d_in[0];   // [N, 64] f32
    const float* W    = (const float*)d_in[1];   // [64, 64] f32
    const float* bias = (const float*)d_in[2];   // [64] f32
    const int*   ei   = (const int*)d_in[3];     // [2, E] int32

    float* out = (float*)d_out;                  // [N, 64] f32

    const int n = in_sizes[0] / D_DIM;           // 100000
    const int e = in_sizes[3] / 2;               // 800000

    // workspace layout: dinv [n], h [n*64]  (both f32; 16B-aligned split)
    float* deg = (float*)d_ws;                   // reused as dinv after k_dinv
    float* h   = deg + n;

    k_init_deg<<<(n + 255) / 256, 256, 0, stream>>>(deg, n);
    k_deg<<<(e + 255) / 256, 256, 0, stream>>>(ei, deg, e);
    k_dinv<<<(n + 255) / 256, 256, 0, stream>>>(deg, n);

    const int strips = (n + 15) / 16;            // 6250
    k_gemm<<<(strips + 7) / 8, 256, 0, stream>>>(x, W, bias, deg, h, out, n);

    const long long ethreads = (long long)e * 16;
    k_edge<<<(int)((ethreads + 255) / 256), 256, 0, stream>>>(ei, h, deg, out, e);

    k_final<<<(n * D_DIM / 4 + 255) / 256, 256, 0, stream>>>(x, out, n * D_DIM / 4);
}